// Track_18494129177154
// MI455X (gfx1250) — compile-verified
//
#include <hip/hip_runtime.h>

// Particle tracker: RK4 in a planar undulator (B = [0, B0*cos(2*pi*z), 0]).
// One thread = one particle; full state in registers across all 1000 steps.
// Output [2, N, 3, T] fp32 staged through LDS in CHUNK-sized time tiles,
// drained with CDNA5 async LDS->global stores (ASYNCcnt) so HBM sees
// 64B-contiguous quads that merge to full lines in L2 across chunks.

#define BLOCK 128
#define CHUNK 16                    // time samples staged per chunk
#define ROWF  (6 * CHUNK + 1)       // per-thread LDS region (floats); +1 pad -> conflict-free banks
#define BUFSZ (BLOCK * ROWF)        // one staging buffer (floats)
#define QPT   (6 * CHUNK / 4)       // b128 quads per thread per chunk (= 24)

__device__ __forceinline__ float cos2pi(float x) { return __builtin_amdgcn_cosf(x); }  // v_cos_f32: cos(2*pi*x)
__device__ __forceinline__ float rsqf(float x)   { return __builtin_amdgcn_rsqf(x); }  // v_rsq_f32
__device__ __forceinline__ float fsqrt(float x)  { return __builtin_amdgcn_sqrtf(x); } // raw v_sqrt_f32

struct Stage { float rkx, rky, rkz, pkx, pkz; };

__device__ __forceinline__ Stage stage(float z, float px, float py, float pz,
                                       float b0, float ic2) {
  // dr/dt = p/gamma ; dp/dt = -cross(p,B)/gamma = (pz*By, 0, -px*By)/gamma
  float By = b0 * cos2pi(z);
  float s2 = fmaf(px, px, fmaf(py, py, pz * pz));
  float ig = rsqf(fmaf(s2, ic2, 1.0f));   // 1/gamma
  Stage o;
  o.rkx = px * ig; o.rky = py * ig; o.rkz = pz * ig;
  float Bg = By * ig;
  o.pkx = pz * Bg; o.pkz = -px * Bg;
  return o;
}

__global__ __launch_bounds__(BLOCK)
void Track_18494129177154_kernel(const float* __restrict__ r_in,
                                 const float* __restrict__ d_dir,
                                 const float* __restrict__ gam,
                                 const float* __restrict__ time,
                                 const float* __restrict__ B0p,
                                 float* __restrict__ out,
                                 int N, int T) {
  __shared__ float smem[2 * BUFSZ];

  const int tid   = threadIdx.x;
  const int pbase = blockIdx.x * BLOCK;
  const int gid   = pbase + tid;

  const float C    = 0.299792458f;
  const float IC2  = 1.0f / (C * C);
  const float INVC = 1.0f / C;
  const float dt   = time[1] - time[0];
  const float dt2  = 0.5f * dt;
  const float dt6  = dt * (1.0f / 6.0f);
  const float b0   = B0p[0];

  // p0 = C*sqrt(gamma^2 - 1) * d/|d|   (gamma ~ 100: no denorm risk, raw v_sqrt is exact enough)
  float rx = 0.f, ry = 0.f, rz = 0.f, px = 0.f, py = 0.f, pz = 1.f;
  if (gid < N) {
    rx = r_in[3 * gid + 0]; ry = r_in[3 * gid + 1]; rz = r_in[3 * gid + 2];
    float dx = d_dir[3 * gid + 0], dy = d_dir[3 * gid + 1], dz = d_dir[3 * gid + 2];
    float g  = gam[gid];
    float sc = C * fsqrt(fmaf(g, g, -1.0f)) * rsqf(fmaf(dx, dx, fmaf(dy, dy, dz * dz)));
    px = dx * sc; py = dy * sc; pz = dz * sc;
  }

  const int nchunk = (T + CHUNK - 1) / CHUNK;
  for (int chunk = 0; chunk < nchunk; ++chunk) {
    float* buf = &smem[(chunk & 1) * BUFSZ];

    // Before re-filling a buffer, drain the async stores that read it two
    // chunks ago. Stores complete in order, so asynccnt<=QPT means only the
    // most recent chunk's QPT stores can still be in flight.
    if (chunk >= 2) {
      asm volatile("s_wait_asynccnt 24" ::: "memory");
      __syncthreads();
    }

    float* myrow = buf + tid * ROWF;
    const int t0 = chunk * CHUNK;
    const int nt = min(CHUNK, T - t0);

    for (int tl = 0; tl < nt; ++tl) {
      Stage k1 = stage(rz, px, py, pz, b0, IC2);
      // sample t: r and beta = p/sqrt(C^2+|p|^2) = (p/gamma) * (1/C) = rk1/C
      myrow[0 * CHUNK + tl] = rx;
      myrow[1 * CHUNK + tl] = ry;
      myrow[2 * CHUNK + tl] = rz;
      myrow[3 * CHUNK + tl] = k1.rkx * INVC;
      myrow[4 * CHUNK + tl] = k1.rky * INVC;
      myrow[5 * CHUNK + tl] = k1.rkz * INVC;

      Stage k2 = stage(fmaf(k1.rkz, dt2, rz), fmaf(k1.pkx, dt2, px), py,
                       fmaf(k1.pkz, dt2, pz), b0, IC2);
      Stage k3 = stage(fmaf(k2.rkz, dt2, rz), fmaf(k2.pkx, dt2, px), py,
                       fmaf(k2.pkz, dt2, pz), b0, IC2);
      Stage k4 = stage(fmaf(k3.rkz, dt,  rz), fmaf(k3.pkx, dt,  px), py,
                       fmaf(k3.pkz, dt,  pz), b0, IC2);

      rx = fmaf(dt6, k1.rkx + 2.0f * (k2.rkx + k3.rkx) + k4.rkx, rx);
      ry = fmaf(dt6, k1.rky + 2.0f * (k2.rky + k3.rky) + k4.rky, ry);
      rz = fmaf(dt6, k1.rkz + 2.0f * (k2.rkz + k3.rkz) + k4.rkz, rz);
      px = fmaf(dt6, k1.pkx + 2.0f * (k2.pkx + k3.pkx) + k4.pkx, px);
      pz = fmaf(dt6, k1.pkz + 2.0f * (k2.pkz + k3.pkz) + k4.pkz, pz);
      // py is exactly conserved by the field geometry
    }
    __syncthreads();

    // Drain this chunk with async LDS->global stores. Lanes cover consecutive
    // t of each (particle, {r,beta}, comp) row.
    const unsigned bufBase = (unsigned)(size_t)(const void*)buf; // low 32b = LDS offset
    const float*   gbase   = out + t0;

    if (t0 + CHUNK <= T) {
      // Fast path (62 of 63 chunks): branch-free unconditional b128 drain.
      for (int i = 0; i < QPT; ++i) {
        int q   = tid + i * BLOCK;
        int row = q >> 2;              // 0 .. BLOCK*6-1
        int k   = q & 3;               // quad within row
        int tr  = row / 6;             // particle slot in block
        int r6  = row - tr * 6;        // 0..2 = r.xyz, 3..5 = beta.xyz
        int s   = (r6 >= 3) ? 1 : 0;
        int c   = r6 - 3 * s;
        int gn  = pbase + tr;
        if (gn >= N) continue;         // never taken when N % BLOCK == 0
        unsigned laddr = bufBase + (unsigned)((tr * ROWF + r6 * CHUNK + 4 * k) * 4);
        const float* g = gbase + (((size_t)s * N + gn) * 3 + c) * (size_t)T + 4 * k;
        asm volatile("global_store_async_from_lds_b128 %0, %1, off"
                     :: "v"(g), "v"(laddr) : "memory");
      }
    } else {
      // Ragged tail chunk (T % CHUNK != 0): per-element b32 stores.
      for (int i = 0; i < QPT; ++i) {
        int q   = tid + i * BLOCK;
        int row = q >> 2;
        int k   = q & 3;
        int tr  = row / 6;
        int r6  = row - tr * 6;
        int s   = (r6 >= 3) ? 1 : 0;
        int c   = r6 - 3 * s;
        int gn  = pbase + tr;
        if (gn >= N) continue;
        unsigned laddr = bufBase + (unsigned)((tr * ROWF + r6 * CHUNK + 4 * k) * 4);
        const float* g = gbase + (((size_t)s * N + gn) * 3 + c) * (size_t)T + 4 * k;
        for (int j = 0; j < 4; ++j) {
          if (t0 + 4 * k + j < T) {
            asm volatile("global_store_async_from_lds_b32 %0, %1, off"
                         :: "v"(g + j), "v"(laddr + 4u * j) : "memory");
          }
        }
      }
    }
  }
  asm volatile("s_wait_asynccnt 0" ::: "memory");
}

extern "C" void kernel_launch(void* const* d_in, const int* in_sizes, int n_in,
                              void* d_out, int out_size, void* d_ws, size_t ws_size,
                              hipStream_t stream) {
  const float* r   = (const float*)d_in[0];   // bunch_r     [N,3]
  const float* d   = (const float*)d_in[1];   // bunch_d     [N,3]
  const float* gam = (const float*)d_in[2];   // bunch_gamma [N]
  const float* t   = (const float*)d_in[3];   // time        [T]
  const float* B0  = (const float*)d_in[4];   // B0          [1]
  float* out = (float*)d_out;                 // [2, N, 3, T] fp32

  const int N = in_sizes[2];
  const int T = in_sizes[3];
  const int grid = (N + BLOCK - 1) / BLOCK;
  Track_18494129177154_kernel<<<grid, BLOCK, 0, stream>>>(r, d, gam, t, B0, out, N, T);
}